// vqvae_85023172591770
// MI455X (gfx1250) — compile-verified
//
#include <hip/hip_runtime.h>
#include <math.h>

// ---------------------------------------------------------------------------
// VQ-VAE forward for gfx1250 (MI455X). Every GEMM-shaped stage (all convs,
// both transpose-conv parities, and the VQ distance search) runs on
// V_WMMA_F32_16X16X4_F32 (exact fp32 matrix path), wave32 layouts per CDNA5
// ISA 7.12.2. Conv waves compute a 16(M) x 64(N) slab: one repacked A
// (weight) fragment = a single aligned b64 load feeding 4 WMMAs per K-step.
// Interior tiles take a check-free loop (wave-uniform dispatch); boundary
// tiles use clamp+select (no EXEC divergence around WMMA).
// ---------------------------------------------------------------------------

typedef __attribute__((ext_vector_type(2))) float v2f;
typedef __attribute__((ext_vector_type(8))) float v8f;

#define BATCH 64
#define HCH   128
#define RHCH  32
#define DLAT  64
#define KCB   512
#define L0    8192
#define L1    4096
#define LW    2048          // latent length
#define NVEC  (BATCH * LW)  // 131072 latent vectors
#define BETA  0.25f

#define WMMA_F32(a, b, c) \
    __builtin_amdgcn_wmma_f32_16x16x4_f32(false, (a), false, (b), (short)0, (c), false, false)

// ---------------------------------------------------------------------------
// Weight repack: wr[co][j] = w[co][ci][k],  j = k*Cin + ci  (GEMM-K order).
// Makes the per-K-step A fragment one contiguous, 8B-aligned v2f load.
// ---------------------------------------------------------------------------
template<int KW, int CIN_LOG2>
__global__ void repack_conv_w(const float* __restrict__ w, float* __restrict__ wr,
                              int total)
{
    constexpr int CIN  = 1 << CIN_LOG2;
    constexpr int KDIM = CIN * KW;
    int i = blockIdx.x * 256 + threadIdx.x;
    if (i >= total) return;
    int co = i / KDIM, j = i - co * KDIM;
    int ci = j & (CIN - 1), k = j >> CIN_LOG2;
    wr[i] = w[(size_t)co * KDIM + ci * KW + k];
}

// dt1 repack: wr[p][o][j], j = t*128+ci; taps per parity: p0 -> {3,1}, p1 -> {2,0}.
__global__ void repack_convT_w(const float* __restrict__ w, float* __restrict__ wr)
{
    int i = blockIdx.x * 256 + threadIdx.x;     // over 2*64*256
    if (i >= 2 * 64 * 256) return;
    int p   = i >> 14;
    int rem = i & 16383;
    int o   = rem >> 8;
    int j   = rem & 255;
    int t   = j >> 7, ci = j & 127;
    int k   = (p == 0) ? (t ? 1 : 3) : (t ? 0 : 2);
    wr[i] = w[(size_t)(ci * 64 + o) * 4 + k];
}

// ---------------------------------------------------------------------------
// Conv inner loop: 4 WMMAs per K-step sharing one A fragment.
// ---------------------------------------------------------------------------
template<int KW, int STRIDE, int PAD, int CIN_LOG2, bool RELU_IN, bool CHECK>
__device__ __forceinline__
void conv_accum(const float* __restrict__ wrow, const float* __restrict__ inb,
                int Lin, int l0, int half, v8f& c0, v8f& c1, v8f& c2, v8f& c3)
{
    constexpr int CIN  = 1 << CIN_LOG2;
    constexpr int KDIM = CIN * KW;
    for (int j0 = 0; j0 < KDIM; j0 += 4) {
        const int jA  = j0 + half * 2;
        const int ci0 = jA & (CIN - 1),       k0 = jA >> CIN_LOG2;
        const int ci1 = (jA + 1) & (CIN - 1), k1 = (jA + 1) >> CIN_LOG2;

        v2f a = *(const v2f*)(wrow + jA);           // repacked: contiguous

        const int q0 = STRIDE * l0 + k0 - PAD;
        const int q1 = STRIDE * l0 + k1 - PAD;
        const float* r0 = inb + (size_t)ci0 * Lin;
        const float* r1 = inb + (size_t)ci1 * Lin;

#pragma unroll
        for (int s = 0; s < 4; ++s) {
            int p0 = q0 + STRIDE * 16 * s;
            int p1 = q1 + STRIDE * 16 * s;
            float x0, x1;
            if (!CHECK) {
                x0 = r0[p0];
                x1 = r1[p1];
            } else {                                // clamp+select, no divergence
                int p0c = p0 < 0 ? 0 : (p0 >= Lin ? Lin - 1 : p0);
                int p1c = p1 < 0 ? 0 : (p1 >= Lin ? Lin - 1 : p1);
                x0 = r0[p0c];
                x1 = r1[p1c];
                x0 = (p0 == p0c) ? x0 : 0.0f;
                x1 = (p1 == p1c) ? x1 : 0.0f;
            }
            if (RELU_IN) { x0 = fmaxf(x0, 0.0f); x1 = fmaxf(x1, 0.0f); }
            v2f bv; bv.x = x0; bv.y = x1;
            if      (s == 0) c0 = WMMA_F32(a, bv, c0);
            else if (s == 1) c1 = WMMA_F32(a, bv, c1);
            else if (s == 2) c2 = WMMA_F32(a, bv, c2);
            else             c3 = WMMA_F32(a, bv, c3);
        }
    }
}

// ---------------------------------------------------------------------------
// Generic conv1d as implicit GEMM, weights pre-repacked to j-order.
// One wave -> 16x64 outputs.
// ---------------------------------------------------------------------------
template<int KW, int STRIDE, int PAD, int CIN_LOG2,
         bool RELU_IN, bool HAS_BIAS, bool HAS_RESID, bool RELU_OUT>
__global__ __launch_bounds__(128)
void conv1d_wmma(const float* __restrict__ in, const float* __restrict__ wr,
                 const float* __restrict__ bias, const float* __restrict__ resid,
                 float* __restrict__ out,
                 int B, int Lin, int Cout, int Lout)
{
    constexpr int CIN  = 1 << CIN_LOG2;
    constexpr int KDIM = CIN * KW;

    const int Mtiles = Cout >> 4;
    const int Ngrp   = Lout >> 6;
    const int total  = B * Mtiles * Ngrp;

    int wid = blockIdx.x * 4 + threadIdx.y;
    if (wid >= total) return;                       // wave-uniform

    int b   = wid / (Mtiles * Ngrp);
    int rem = wid - b * (Mtiles * Ngrp);
    int mt  = rem / Ngrp;
    int ng  = rem - mt * Ngrp;

    const int lane  = threadIdx.x;
    const int half  = lane >> 4;
    const int lanem = lane & 15;
    const int l0    = ng * 64 + lanem;

    const float* wrow = wr + (size_t)(mt * 16 + lanem) * KDIM;
    const float* inb  = in + (size_t)b * CIN * Lin;

    v8f c0 = {}, c1 = {}, c2 = {}, c3 = {};

    // wave-uniform interior test: columns ng*64 .. ng*64+63, taps 0..KW-1
    bool interior = (PAD == 0) ||
                    ((STRIDE * (ng * 64) - PAD >= 0) &&
                     (STRIDE * (ng * 64 + 63) + (KW - 1) - PAD < Lin));
    if (interior)
        conv_accum<KW, STRIDE, PAD, CIN_LOG2, RELU_IN, false>(wrow, inb, Lin, l0, half, c0, c1, c2, c3);
    else
        conv_accum<KW, STRIDE, PAD, CIN_LOG2, RELU_IN, true >(wrow, inb, Lin, l0, half, c0, c1, c2, c3);

#pragma unroll
    for (int s = 0; s < 4; ++s) {
        const v8f& c = (s == 0) ? c0 : (s == 1) ? c1 : (s == 2) ? c2 : c3;
#pragma unroll
        for (int r = 0; r < 8; ++r) {
            int co = mt * 16 + r + half * 8;
            size_t oidx = ((size_t)b * Cout + co) * Lout + l0 + 16 * s;
            float v = c[r];
            if (HAS_BIAS)  v += bias[co];
            if (HAS_RESID) v += resid[oidx];
            if (RELU_OUT)  v  = fmaxf(v, 0.0f);
            out[oidx] = v;
        }
    }
}

// ---------------------------------------------------------------------------
// dt1 ConvTranspose1d (s2,p1,k4, 128->64) on WMMA; 2-tap GEMM per parity.
//   even t=2m:  sum_i w[i,o,3]*X[i,m-1] + w[i,o,1]*X[i,m]
//   odd  t=2m+1: sum_i w[i,o,2]*X[i,m]   + w[i,o,0]*X[i,m+1]
// Input ReLU (decoder stack tail), output ReLU. Weights pre-repacked.
// ---------------------------------------------------------------------------
template<int DP, bool CHECK>
__device__ __forceinline__
void convT_accum(const float* __restrict__ wrow, const float* __restrict__ inb,
                 int m0, int half, v8f& c0, v8f& c1, v8f& c2, v8f& c3)
{
    for (int j0 = 0; j0 < 256; j0 += 4) {
        const int jA  = j0 + half * 2;
        const int ci0 = jA & 127,       t0 = jA >> 7;
        const int ci1 = (jA + 1) & 127, t1 = (jA + 1) >> 7;

        v2f a = *(const v2f*)(wrow + jA);

        const int q0 = m0 + t0 + DP;
        const int q1 = m0 + t1 + DP;
        const float* r0 = inb + (size_t)ci0 * LW;
        const float* r1 = inb + (size_t)ci1 * LW;

#pragma unroll
        for (int s = 0; s < 4; ++s) {
            int p0 = q0 + 16 * s;
            int p1 = q1 + 16 * s;
            float x0, x1;
            if (!CHECK) {
                x0 = r0[p0];
                x1 = r1[p1];
            } else {
                int p0c = p0 < 0 ? 0 : (p0 >= LW ? LW - 1 : p0);
                int p1c = p1 < 0 ? 0 : (p1 >= LW ? LW - 1 : p1);
                x0 = r0[p0c];
                x1 = r1[p1c];
                x0 = (p0 == p0c) ? x0 : 0.0f;
                x1 = (p1 == p1c) ? x1 : 0.0f;
            }
            x0 = fmaxf(x0, 0.0f);
            x1 = fmaxf(x1, 0.0f);
            v2f bv; bv.x = x0; bv.y = x1;
            if      (s == 0) c0 = WMMA_F32(a, bv, c0);
            else if (s == 1) c1 = WMMA_F32(a, bv, c1);
            else if (s == 2) c2 = WMMA_F32(a, bv, c2);
            else             c3 = WMMA_F32(a, bv, c3);
        }
    }
}

template<int PARITY>
__global__ __launch_bounds__(128)
void convT1_wmma(const float* __restrict__ in,   // [B, 128, LW]
                 const float* __restrict__ wrp,  // repacked [2][64][256]
                 const float* __restrict__ bias, // [64]
                 float* __restrict__ out)        // [B, 64, L1]
{
    constexpr int DP = (PARITY == 0) ? -1 : 0;

    const int Mtiles = 4;
    const int Ngrp   = LW >> 6;                 // 32
    const int total  = BATCH * Mtiles * Ngrp;

    int wid = blockIdx.x * 4 + threadIdx.y;
    if (wid >= total) return;

    int b   = wid / (Mtiles * Ngrp);
    int rem = wid - b * (Mtiles * Ngrp);
    int mt  = rem / Ngrp;
    int ng  = rem - mt * Ngrp;

    const int lane  = threadIdx.x;
    const int half  = lane >> 4;
    const int lanem = lane & 15;
    const int m0    = ng * 64 + lanem;

    const float* wrow = wrp + (size_t)PARITY * 16384 + (size_t)(mt * 16 + lanem) * 256;
    const float* inb  = in + (size_t)b * HCH * LW;

    v8f c0 = {}, c1 = {}, c2 = {}, c3 = {};

    bool interior = (ng * 64 + DP >= 0) && (ng * 64 + 63 + 1 + DP < LW);
    if (interior) convT_accum<DP, false>(wrow, inb, m0, half, c0, c1, c2, c3);
    else          convT_accum<DP, true >(wrow, inb, m0, half, c0, c1, c2, c3);

#pragma unroll
    for (int s = 0; s < 4; ++s) {
        const v8f& c = (s == 0) ? c0 : (s == 1) ? c1 : (s == 2) ? c2 : c3;
#pragma unroll
        for (int r = 0; r < 8; ++r) {
            int co = mt * 16 + r + half * 8;
            int t  = 2 * (m0 + 16 * s) + PARITY;
            out[((size_t)b * 64 + co) * L1 + t] = fmaxf(c[r] + bias[co], 0.0f);
        }
    }
}

// ---------------------------------------------------------------------------
// VQ on WMMA: wave owns 16 latent vectors (B fragments preloaded to 32 VGPRs),
// sweeps 32 codebook tiles x 16 WMMAs. argmin over (||c||^2 - 2 c.z) per lane
// with index tie-break; halves merged via shfl_xor(16). Lanes 0-15 then
// recompute exact ||z-c||^2, write q over z in place, bump counts.
// ---------------------------------------------------------------------------
__global__ __launch_bounds__(128)
void vq_wmma(float* __restrict__ z,              // [B, D, LW] (q in place)
             const float* __restrict__ cb,       // [K, D]
             const float* __restrict__ cbn,      // [K]  ||c||^2
             float* __restrict__ mse,            // [NVEC]
             float* __restrict__ counts)         // [K]
{
    const int Ngrp = LW >> 4;                    // 128 column groups / batch
    int wid = blockIdx.x * 4 + threadIdx.y;      // 8192 waves
    int b   = wid / Ngrp;
    int gt  = wid - b * Ngrp;

    const int lane  = threadIdx.x;
    const int half  = lane >> 4;
    const int lanem = lane & 15;
    const int col   = gt * 16 + lanem;

    float* zcol = z + (size_t)b * DLAT * LW + col;

    v2f zf[16];
#pragma unroll
    for (int st = 0; st < 16; ++st) {
        int d = 4 * st + 2 * half;
        v2f t;
        t.x = zcol[(size_t)d * LW];
        t.y = zcol[(size_t)(d + 1) * LW];
        zf[st] = t;
    }

    float best = 3.4e38f;
    int   bid  = 0;
    for (int mt = 0; mt < 32; ++mt) {
        const float* crow = cb + (size_t)(mt * 16 + lanem) * DLAT + 2 * half;
        v8f acc = {};
#pragma unroll
        for (int st = 0; st < 16; ++st) {
            v2f a = *(const v2f*)(crow + 4 * st);
            acc = WMMA_F32(a, zf[st], acc);
        }
#pragma unroll
        for (int r = 0; r < 8; ++r) {
            int code = mt * 16 + 8 * half + r;
            float dist = cbn[code] - 2.0f * acc[r];
            if (dist < best) { best = dist; bid = code; }
        }
    }

    float ob = __shfl_xor(best, 16, 32);
    int   oi = __shfl_xor(bid,  16, 32);
    if (ob < best || (ob == best && oi < bid)) { best = ob; bid = oi; }

    if (half == 0) {                             // one lane per column
        const float* cq = cb + (size_t)bid * DLAT;
        float s = 0.0f;
        for (int d = 0; d < DLAT; ++d) {
            float zv = zcol[(size_t)d * LW];
            float cv = cq[d];
            float df = zv - cv;
            s = fmaf(df, df, s);
            zcol[(size_t)d * LW] = cv;           // q over z
        }
        mse[b * LW + col] = s;
        atomicAdd(&counts[bid], 1.0f);           // exact integer sums
    }
}

__global__ void cb_norm_kernel(const float* __restrict__ cb, float* __restrict__ cbn)
{
    int k = blockIdx.x * 256 + threadIdx.x;
    if (k >= KCB) return;
    float s = 0.0f;
    for (int d = 0; d < DLAT; ++d) { float v = cb[k * DLAT + d]; s = fmaf(v, v, s); }
    cbn[k] = s;
}

// dt2: ConvTranspose1d 64 -> 1, s2, p1, k4. M=1: VALU.
__global__ __launch_bounds__(256)
void convT2_kernel(const float* __restrict__ in,    // [B, 64, L1]
                   const float* __restrict__ w,     // [64, 1, 4]
                   const float* __restrict__ bias,  // [1]
                   float* __restrict__ out)         // [B, L0]
{
    int tid = blockIdx.x * 256 + threadIdx.x;
    int t = tid & (L0 - 1);
    int b = tid >> 13;
    int m = t >> 1;
    bool even = (t & 1) == 0;
    int la = even ? m     : m + 1;  int kA = even ? 1 : 0;
    int lb = even ? m - 1 : m;      int kB = even ? 3 : 2;
    bool va = (la >= 0 && la < L1);
    bool vb = (lb >= 0 && lb < L1);
    int lac = va ? la : 0;
    int lbc = vb ? lb : 0;

    const float* inb = in + (size_t)b * 64 * L1;
    float acc = bias[0];
    for (int i = 0; i < 64; ++i) {
        float xa = inb[(size_t)i * L1 + lac]; xa = va ? xa : 0.0f;
        float xb = inb[(size_t)i * L1 + lbc]; xb = vb ? xb : 0.0f;
        acc = fmaf(xa, w[i * 4 + kA], acc);
        acc = fmaf(xb, w[i * 4 + kB], acc);
    }
    out[tid] = acc;
}

__global__ void zero_kernel(float* __restrict__ p, int n)
{
    int i = blockIdx.x * 256 + threadIdx.x;
    if (i < n) p[i] = 0.0f;
}

// Deterministic 256 -> 1 tree reduction per block.
__global__ __launch_bounds__(256)
void reduce256_kernel(const float* __restrict__ src, float* __restrict__ dst)
{
    __shared__ float s[256];
    int t = threadIdx.x;
    s[t] = src[blockIdx.x * 256 + t];
    __syncthreads();
    for (int k = 128; k > 0; k >>= 1) {
        if (t < k) s[t] += s[t + k];
        __syncthreads();
    }
    if (t == 0) dst[blockIdx.x] = s[0];
}

__global__ void finalize_kernel(const float* __restrict__ partial, int nPart,
                                const float* __restrict__ counts,
                                float* __restrict__ out2)
{
    if (threadIdx.x != 0 || blockIdx.x != 0) return;
    float mse = 0.0f;
    for (int i = 0; i < nPart; ++i) mse += partial[i];           // fixed order
    float mean = mse / ((float)NVEC * (float)DLAT);
    out2[0] = (1.0f + BETA) * mean;                               // vq_loss

    const float invN = 1.0f / (float)NVEC;
    float H = 0.0f;
    for (int k = 0; k < KCB; ++k) {
        float p = counts[k] * invN;
        H -= p * logf(p + 1e-10f);
    }
    out2[1] = expf(H);                                            // perplexity
}

// ---------------------------------------------------------------------------
// Host orchestration
// ---------------------------------------------------------------------------
static inline int slab_blocks(int B, int Cout, int Lout)
{
    int waves = B * (Cout >> 4) * (Lout >> 6);
    return (waves + 3) / 4;
}

extern "C" void kernel_launch(void* const* d_in, const int* in_sizes, int n_in,
                              void* d_out, int out_size, void* d_ws, size_t ws_size,
                              hipStream_t stream)
{
    (void)in_sizes; (void)n_in; (void)out_size; (void)ws_size;

    const float* x      = (const float*)d_in[0];
    const float* ec1_w  = (const float*)d_in[1];
    const float* ec1_b  = (const float*)d_in[2];
    const float* ec2_w  = (const float*)d_in[3];
    const float* ec2_b  = (const float*)d_in[4];
    const float* ec3_w  = (const float*)d_in[5];
    const float* ec3_b  = (const float*)d_in[6];
    const float* er1_w1 = (const float*)d_in[7];
    const float* er1_w2 = (const float*)d_in[8];
    const float* er2_w1 = (const float*)d_in[9];
    const float* er2_w2 = (const float*)d_in[10];
    const float* pvq_w  = (const float*)d_in[11];
    const float* pvq_b  = (const float*)d_in[12];
    const float* cbook  = (const float*)d_in[13];
    const float* dc1_w  = (const float*)d_in[14];
    const float* dc1_b  = (const float*)d_in[15];
    const float* dr1_w1 = (const float*)d_in[16];
    const float* dr1_w2 = (const float*)d_in[17];
    const float* dr2_w1 = (const float*)d_in[18];
    const float* dr2_w2 = (const float*)d_in[19];
    const float* dt1_w  = (const float*)d_in[20];
    const float* dt1_b  = (const float*)d_in[21];
    const float* dt2_w  = (const float*)d_in[22];
    const float* dt2_b  = (const float*)d_in[23];

    float* out = (float*)d_out;   // [recon(524288) | vq_loss | perplexity]
    float* ws  = (float*)d_ws;

    // ---- workspace layout (floats) ----
    const size_t SZ_A = (size_t)BATCH * 64  * L1;   // ec1 out -> dt1 out
    const size_t SZ_B = (size_t)BATCH * HCH * LW;   // ec2 out -> decoder trunk
    const size_t SZ_C = (size_t)BATCH * HCH * LW;   // encoder trunk
    const size_t SZ_T = (size_t)BATCH * RHCH * LW;  // residual hidden
    const size_t SZ_Z = (size_t)BATCH * DLAT * LW;  // z (q in place)
    float* A  = ws;
    float* Bb = A  + SZ_A;
    float* C  = Bb + SZ_B;
    float* T  = C  + SZ_C;
    float* Z  = T  + SZ_T;
    float* P  = Z  + SZ_Z;
    // repacked weights (KW>1 layers only; KW==1 layouts are already j-ordered)
    float* wr_ec1  = P;             P += 64  * 4;       // 256
    float* wr_ec2  = P;             P += 128 * 256;     // 32768
    float* wr_ec3  = P;             P += 128 * 384;     // 49152
    float* wr_er1  = P;             P += 32  * 384;     // 12288
    float* wr_er2  = P;             P += 32  * 384;
    float* wr_dc1  = P;             P += 128 * 192;     // 24576
    float* wr_dr1  = P;             P += 32  * 384;
    float* wr_dr2  = P;             P += 32  * 384;
    float* wr_dt1  = P;             P += 2 * 64 * 256;  // 32768
    float* CBN     = P;             P += KCB;           // codebook norms
    float* CNT     = P;             P += KCB;           // histogram
    float* PART    = P;             P += NVEC / 256;    // 512 partials
    float* MSE     = P;             P += NVEC;          // per-vector mse

    const dim3 cblk(32, 4);

    // ---- one-shot preprocessing (tiny) ----
    repack_conv_w<4, 0><<<1, 256, 0, stream>>>(ec1_w, wr_ec1, 64 * 4);
    repack_conv_w<4, 6><<<128, 256, 0, stream>>>(ec2_w, wr_ec2, 128 * 256);
    repack_conv_w<3, 7><<<192, 256, 0, stream>>>(ec3_w, wr_ec3, 128 * 384);
    repack_conv_w<3, 7><<<48, 256, 0, stream>>>(er1_w1, wr_er1, 32 * 384);
    repack_conv_w<3, 7><<<48, 256, 0, stream>>>(er2_w1, wr_er2, 32 * 384);
    repack_conv_w<3, 6><<<96, 256, 0, stream>>>(dc1_w, wr_dc1, 128 * 192);
    repack_conv_w<3, 7><<<48, 256, 0, stream>>>(dr1_w1, wr_dr1, 32 * 384);
    repack_conv_w<3, 7><<<48, 256, 0, stream>>>(dr2_w1, wr_dr2, 32 * 384);
    repack_convT_w<<<128, 256, 0, stream>>>(dt1_w, wr_dt1);
    cb_norm_kernel<<<2, 256, 0, stream>>>(cbook, CBN);
    zero_kernel<<<2, 256, 0, stream>>>(CNT, KCB);

    // ---- Encoder ----
    conv1d_wmma<4, 2, 1, 0, false, true, false, true>
        <<<slab_blocks(BATCH, 64, L1), cblk, 0, stream>>>(
            x, wr_ec1, ec1_b, nullptr, A, BATCH, L0, 64, L1);
    conv1d_wmma<4, 2, 1, 6, false, true, false, true>
        <<<slab_blocks(BATCH, HCH, LW), cblk, 0, stream>>>(
            A, wr_ec2, ec2_b, nullptr, Bb, BATCH, L1, HCH, LW);
    conv1d_wmma<3, 1, 1, 7, false, true, false, false>
        <<<slab_blocks(BATCH, HCH, LW), cblk, 0, stream>>>(
            Bb, wr_ec3, ec3_b, nullptr, C, BATCH, LW, HCH, LW);

    // residual blocks (KW==1 weights used unrepacked: identical layout)
    conv1d_wmma<3, 1, 1, 7, true, false, false, false>
        <<<slab_blocks(BATCH, RHCH, LW), cblk, 0, stream>>>(
            C, wr_er1, nullptr, nullptr, T, BATCH, LW, RHCH, LW);
    conv1d_wmma<1, 1, 0, 5, true, false, true, false>
        <<<slab_blocks(BATCH, HCH, LW), cblk, 0, stream>>>(
            T, er1_w2, nullptr, C, C, BATCH, LW, HCH, LW);
    conv1d_wmma<3, 1, 1, 7, true, false, false, false>
        <<<slab_blocks(BATCH, RHCH, LW), cblk, 0, stream>>>(
            C, wr_er2, nullptr, nullptr, T, BATCH, LW, RHCH, LW);
    conv1d_wmma<1, 1, 0, 5, true, false, true, false>
        <<<slab_blocks(BATCH, HCH, LW), cblk, 0, stream>>>(
            T, er2_w2, nullptr, C, C, BATCH, LW, HCH, LW);
    conv1d_wmma<1, 1, 0, 7, true, true, false, false>
        <<<slab_blocks(BATCH, DLAT, LW), cblk, 0, stream>>>(
            C, pvq_w, pvq_b, nullptr, Z, BATCH, LW, DLAT, LW);

    // ---- Vector quantizer (WMMA distance GEMM + argmin) ----
    vq_wmma<<<(NVEC / 16 + 3) / 4, cblk, 0, stream>>>(Z, cbook, CBN, MSE, CNT);
    reduce256_kernel<<<NVEC / 256, 256, 0, stream>>>(MSE, PART);

    // ---- Decoder ----
    conv1d_wmma<3, 1, 1, 6, false, true, false, false>
        <<<slab_blocks(BATCH, HCH, LW), cblk, 0, stream>>>(
            Z, wr_dc1, dc1_b, nullptr, Bb, BATCH, LW, HCH, LW);
    conv1d_wmma<3, 1, 1, 7, true, false, false, false>
        <<<slab_blocks(BATCH, RHCH, LW), cblk, 0, stream>>>(
            Bb, wr_dr1, nullptr, nullptr, T, BATCH, LW, RHCH, LW);
    conv1d_wmma<1, 1, 0, 5, true, false, true, false>
        <<<slab_blocks(BATCH, HCH, LW), cblk, 0, stream>>>(
            T, dr1_w2, nullptr, Bb, Bb, BATCH, LW, HCH, LW);
    conv1d_wmma<3, 1, 1, 7, true, false, false, false>
        <<<slab_blocks(BATCH, RHCH, LW), cblk, 0, stream>>>(
            Bb, wr_dr2, nullptr, nullptr, T, BATCH, LW, RHCH, LW);
    conv1d_wmma<1, 1, 0, 5, true, false, true, false>
        <<<slab_blocks(BATCH, HCH, LW), cblk, 0, stream>>>(
            T, dr2_w2, nullptr, Bb, Bb, BATCH, LW, HCH, LW);

    {
        int blocks = (BATCH * 4 * (LW >> 6) + 3) / 4;
        convT1_wmma<0><<<blocks, cblk, 0, stream>>>(Bb, wr_dt1, dt1_b, A);
        convT1_wmma<1><<<blocks, cblk, 0, stream>>>(Bb, wr_dt1, dt1_b, A);
    }
    convT2_kernel<<<((size_t)BATCH * L0) / 256, 256, 0, stream>>>(
        A, dt2_w, dt2_b, out);

    finalize_kernel<<<1, 32, 0, stream>>>(PART, NVEC / 256, CNT,
                                          out + (size_t)BATCH * L0);
}